// RadialDCLayer_72181220376637
// MI455X (gfx1250) — compile-verified
//
#include <hip/hip_runtime.h>

#define IMG   256
#define GRID  512
#define NJ    6
#define KS    131072      // 256*512 samples
#define NC    8
#define PI_F  3.14159265358979323846f
#define BETA_F 13.8551003f          // pi*sqrt((J/alpha)^2*(alpha-.5)^2-0.8), alpha=2
#define GSZ   (GRID*GRID)           // 262144

typedef float v2f __attribute__((ext_vector_type(2)));
typedef float v8f __attribute__((ext_vector_type(8)));

// ---------------- setup kernels ----------------

__global__ void fill_zero_k(float* p, long long n) {
  long long i = (long long)blockIdx.x * blockDim.x + threadIdx.x;
  if (i < n) p[i] = 0.0f;
}

__global__ void scale1d_k(float* sc) {
  int i = threadIdx.x;                       // 256 threads
  if (i >= IMG) return;
  float n  = (i - IMG * 0.5f) / (float)GRID;
  float t  = BETA_F * BETA_F - (PI_F * NJ * n) * (PI_F * NJ * n);
  float sp = sqrtf(fmaxf(t, 1e-12f));
  float sn = sqrtf(fmaxf(-t, 1e-12f));
  float ft = (t > 0.0f) ? (sinhf(sp) / sp) : (sinf(sn) / sn);
  float sc0 = sinhf(BETA_F) / BETA_F;
  sc[i] = sc0 / ft;
}

__global__ void dft_tables_k(float* Ct, float* St) {
  int t = blockIdx.x * blockDim.x + threadIdx.x;
  if (t >= GSZ) return;
  int i = t >> 9, j = t & (GRID - 1);
  int ph = (i * j) & (GRID - 1);             // exact mod-512 phase
  float th = ph * (2.0f * PI_F / (float)GRID);
  float s, c;
  __sincosf(th, &s, &c);
  Ct[t] = c;  St[t] = s;                     // F = C - i*S (sign applied at use)
}

__device__ __forceinline__ float i0f(float x) {
  float ax = fabsf(x);
  if (ax < 3.75f) {
    float t = (x / 3.75f) * (x / 3.75f);
    return 1.0f + t * (3.5156229f + t * (3.0899424f + t * (1.2067492f +
           t * (0.2659732f + t * (0.0360768f + t * 0.0045813f)))));
  }
  float t = 3.75f / ax;
  return (expf(ax) * rsqrtf(ax)) *
         (0.39894228f + t * (0.01328592f + t * (0.00225319f + t * (-0.00157565f +
          t * (0.00916281f + t * (-0.02057706f + t * (0.02635537f +
          t * (-0.01647633f + t * 0.00392377f))))))));
}

__global__ void coords_k(const float* __restrict__ ktraj,
                         int* __restrict__ iy, int* __restrict__ ix,
                         float* __restrict__ wy, float* __restrict__ wx) {
  int k = blockIdx.x * blockDim.x + threadIdx.x;
  if (k >= KS) return;
  for (int d = 0; d < 2; ++d) {
    float tm   = ktraj[d * KS + k] * ((float)GRID / (2.0f * PI_F));
    float base = floorf(tm - 0.5f * NJ);
    int*   gi = (d == 0) ? iy : ix;
    float* w  = (d == 0) ? wy : wx;
    for (int j = 1; j <= NJ; ++j) {
      float idx = base + (float)j;
      float u   = tm - idx;
      float q   = fmaxf(1.0f - (2.0f * u / NJ) * (2.0f * u / NJ), 0.0f);
      w[k * NJ + j - 1] = i0f(BETA_F * sqrtf(q)) * (1.0f / NJ);
      int ii = (int)idx;
      gi[k * NJ + j - 1] = ((ii % GRID) + GRID) % GRID;
    }
  }
}

// pad = x * smaps * sc2 into 512x512 zero-padded complex grids (8 coils)
__global__ void make_pad_k(const float* __restrict__ ximg,
                           const float* __restrict__ smre, const float* __restrict__ smim,
                           const float* __restrict__ sc,
                           float* __restrict__ padRe, float* __restrict__ padIm) {
  int t = blockIdx.x * blockDim.x + threadIdx.x;
  if (t >= NC * GSZ) return;
  int c = t / GSZ, p = t % GSZ;
  int yy = p >> 9, xx = p & (GRID - 1);
  float re = 0.0f, im = 0.0f;
  if (yy < IMG && xx < IMG) {
    int q = yy * IMG + xx;
    float v = ximg[q] * sc[yy] * sc[xx];
    re = v * smre[c * IMG * IMG + q];
    im = v * smim[c * IMG * IMG + q];
  }
  padRe[t] = re;  padIm[t] = im;
}

// ---------------- WMMA complex GEMM: D = A x B (512x512 complex) ----------------
// A = Are + i*sA*Aim0, B = Bre + i*sB*Bim0.  Four raw WMMA accumulators; the
// complex signs are applied once in the epilogue so the inner loop is pure
// loads + 4x v_wmma_f32_16x16x4_f32 per k-step (no VALU in the chain).
__global__ __launch_bounds__(128)
void cgemm512_k(const float* __restrict__ Are, const float* __restrict__ Aim,
                float sA, long long strideA,
                const float* __restrict__ Bre, const float* __restrict__ Bim,
                float sB, long long strideB,
                float* __restrict__ Dre, float* __restrict__ Dim,
                long long strideD, float scale) {
  const long long coil = blockIdx.z;
  Are += coil * strideA;  Aim += coil * strideA;
  Bre += coil * strideB;  Bim += coil * strideB;
  Dre += coil * strideD;  Dim += coil * strideD;

  const int lane  = threadIdx.x;               // 0..31 (wave32)
  const int mTile = blockIdx.y * 4 + threadIdx.y;
  const int nTile = blockIdx.x;
  const int half  = lane >> 4;                 // 0/1
  const int l15   = lane & 15;
  const int arow  = mTile * 16 + l15;          // A row for this lane
  const int bcol  = nTile * 16 + l15;          // B col for this lane

  v8f accRR = {}; v8f accII = {}; v8f accRI = {}; v8f accIR = {};
  for (int k = 0; k < GRID; k += 4) {
    const int kidx = k + 2 * half;             // A layout: v0={K0,K2}, v1={K1,K3}
    v2f are = *(const v2f*)(Are + (size_t)arow * GRID + kidx);
    v2f aim = *(const v2f*)(Aim + (size_t)arow * GRID + kidx);
    v2f bre, bim;
    bre.x = Bre[(size_t)kidx * GRID + bcol];
    bre.y = Bre[(size_t)(kidx + 1) * GRID + bcol];
    bim.x = Bim[(size_t)kidx * GRID + bcol];
    bim.y = Bim[(size_t)(kidx + 1) * GRID + bcol];
    accRR = __builtin_amdgcn_wmma_f32_16x16x4_f32(false, are, false, bre, (short)0, accRR, false, false);
    accII = __builtin_amdgcn_wmma_f32_16x16x4_f32(false, aim, false, bim, (short)0, accII, false, false);
    accRI = __builtin_amdgcn_wmma_f32_16x16x4_f32(false, are, false, bim, (short)0, accRI, false, false);
    accIR = __builtin_amdgcn_wmma_f32_16x16x4_f32(false, aim, false, bre, (short)0, accIR, false, false);
  }
  // Dre = (accRR - sA*sB*accII)*scale ; Dim = (sB*accRI + sA*accIR)*scale
  const float sAB = sA * sB;
  for (int r = 0; r < 8; ++r) {
    int row = mTile * 16 + r + 8 * half;
    int col = nTile * 16 + l15;
    Dre[(size_t)row * GRID + col] = (accRR[r] - sAB * accII[r]) * scale;
    Dim[(size_t)row * GRID + col] = (sB * accRI[r] + sA * accIR[r]) * scale;
  }
}

// ---------------- gather (forward sample) + DC mix ----------------
__global__ void gather_dc_k(const float* __restrict__ kgRe, const float* __restrict__ kgIm,
                            const int* __restrict__ iy, const int* __restrict__ ix,
                            const float* __restrict__ wy, const float* __restrict__ wx,
                            const float* __restrict__ yre, const float* __restrict__ yim,
                            const float* __restrict__ lraw,
                            float* __restrict__ kdcRe, float* __restrict__ kdcIm) {
  int t = blockIdx.x * blockDim.x + threadIdx.x;
  if (t >= NC * KS) return;
  int c = t >> 17, k = t & (KS - 1);
  const float* gr = kgRe + (size_t)c * GSZ;
  const float* gi = kgIm + (size_t)c * GSZ;
  int   iyv[NJ], ixv[NJ];
  float wyv[NJ], wxv[NJ];
  for (int j = 0; j < NJ; ++j) {
    iyv[j] = iy[k * NJ + j];  ixv[j] = ix[k * NJ + j];
    wyv[j] = wy[k * NJ + j];  wxv[j] = wx[k * NJ + j];
  }
  float ar = 0.0f, ai = 0.0f;
  for (int jy = 0; jy < NJ; ++jy) {
    int roff = iyv[jy] * GRID;
    float wj = wyv[jy];
    for (int jx = 0; jx < NJ; ++jx) {
      float w  = wj * wxv[jx];
      int  idx = roff + ixv[jx];
      ar += gr[idx] * w;
      ai += gi[idx] * w;
    }
  }
  float lam = 1.0f / (1.0f + expf(-lraw[0]));
  kdcRe[t] = lam * ar + (1.0f - lam) * yre[t];
  kdcIm[t] = lam * ai + (1.0f - lam) * yim[t];
}

// ---------------- scatter (adjoint spread) ----------------
__global__ void scatter_adj_k(const float* __restrict__ kdcRe, const float* __restrict__ kdcIm,
                              const int* __restrict__ iy, const int* __restrict__ ix,
                              const float* __restrict__ wy, const float* __restrict__ wx,
                              float* __restrict__ gridRe, float* __restrict__ gridIm) {
  int t = blockIdx.x * blockDim.x + threadIdx.x;
  if (t >= NC * KS) return;
  int c = t >> 17, k = t & (KS - 1);
  float vr = kdcRe[t], vi = kdcIm[t];
  float* gr = gridRe + (size_t)c * GSZ;
  float* gi = gridIm + (size_t)c * GSZ;
  for (int jy = 0; jy < NJ; ++jy) {
    int   roff = iy[k * NJ + jy] * GRID;
    float wj   = wy[k * NJ + jy];
    for (int jx = 0; jx < NJ; ++jx) {
      float w  = wj * wx[k * NJ + jx];
      int  idx = roff + ix[k * NJ + jx];
      atomicAdd(&gr[idx], vr * w);
      atomicAdd(&gi[idx], vi * w);
    }
  }
}

// ---------------- final: crop * sc2, conj(smaps), coil sum ----------------
__global__ void combine_out_k(const float* __restrict__ imgRe, const float* __restrict__ imgIm,
                              const float* __restrict__ smre, const float* __restrict__ smim,
                              const float* __restrict__ sc, float* __restrict__ out) {
  int p = blockIdx.x * blockDim.x + threadIdx.x;
  if (p >= IMG * IMG) return;
  int yy = p >> 8, xx = p & (IMG - 1);
  float s = sc[yy] * sc[xx];
  float ar = 0.0f, ai = 0.0f;
  for (int c = 0; c < NC; ++c) {
    float ir = imgRe[(size_t)c * GSZ + yy * GRID + xx] * s;
    float ii = imgIm[(size_t)c * GSZ + yy * GRID + xx] * s;
    float mr = smre[c * IMG * IMG + p], mi = smim[c * IMG * IMG + p];
    ar += mr * ir + mi * ii;   // conj(smap) * img
    ai += mr * ii - mi * ir;
  }
  out[2 * p]     = ar;         // complex64 interleaved
  out[2 * p + 1] = ai;
}

// ---------------- launcher ----------------
extern "C" void kernel_launch(void* const* d_in, const int* in_sizes, int n_in,
                              void* d_out, int out_size, void* d_ws, size_t ws_size,
                              hipStream_t stream) {
  (void)in_sizes; (void)n_in; (void)out_size; (void)ws_size;
  const float* x_img = (const float*)d_in[0];   // (1,1,256,256)
  const float* y_re  = (const float*)d_in[1];   // (1,8,K)
  const float* y_im  = (const float*)d_in[2];
  const float* sm_re = (const float*)d_in[3];   // (1,8,256,256)
  const float* sm_im = (const float*)d_in[4];
  const float* ktraj = (const float*)d_in[5];   // (2,K)
  const float* lraw  = (const float*)d_in[6];   // (1,)
  float* out = (float*)d_out;

  // workspace carve (floats)
  float* w = (float*)d_ws;
  size_t off = 0;
  float* sc    = w + off; off += 256;
  float* Ct    = w + off; off += GSZ;
  float* St    = w + off; off += GSZ;
  int*   iy    = (int*)(w + off); off += (size_t)KS * NJ;
  int*   ix    = (int*)(w + off); off += (size_t)KS * NJ;
  float* wy    = w + off; off += (size_t)KS * NJ;
  float* wx    = w + off; off += (size_t)KS * NJ;
  float* padRe = w + off; off += (size_t)NC * GSZ;
  float* padIm = w + off; off += (size_t)NC * GSZ;
  float* tmpRe = w + off; off += (size_t)NC * GSZ;
  float* tmpIm = w + off; off += (size_t)NC * GSZ;
  float* kdcRe = w + off; off += (size_t)NC * KS;
  float* kdcIm = w + off; off += (size_t)NC * KS;

  const dim3 gB(256);
  // setup
  scale1d_k<<<1, 256, 0, stream>>>(sc);
  dft_tables_k<<<GSZ / 256, gB, 0, stream>>>(Ct, St);
  coords_k<<<KS / 256, gB, 0, stream>>>(ktraj, iy, ix, wy, wx);
  make_pad_k<<<(NC * GSZ) / 256, gB, 0, stream>>>(x_img, sm_re, sm_im, sc, padRe, padIm);

  // forward FFT via DFT matmuls: kg = F * pad * F / 512   (F = C - iS -> sA/sB = -1)
  dim3 gg(GRID / 16, GRID / 16 / 4, NC), gb(32, 4);
  cgemm512_k<<<gg, gb, 0, stream>>>(Ct, St, -1.0f, 0,
                                    padRe, padIm, 1.0f, (long long)GSZ,
                                    tmpRe, tmpIm, (long long)GSZ, 1.0f);
  cgemm512_k<<<gg, gb, 0, stream>>>(tmpRe, tmpIm, 1.0f, (long long)GSZ,
                                    Ct, St, -1.0f, 0,
                                    padRe, padIm, (long long)GSZ, 1.0f / (float)GRID);

  // gather + DC mix
  gather_dc_k<<<(NC * KS) / 256, gB, 0, stream>>>(padRe, padIm, iy, ix, wy, wx,
                                                  y_re, y_im, lraw, kdcRe, kdcIm);

  // adjoint: zero grids, scatter, inverse DFT (conj(F) = C + iS), *512/512^2 = /512
  fill_zero_k<<<(2 * NC * GSZ) / 256, gB, 0, stream>>>(tmpRe, (long long)2 * NC * GSZ);
  scatter_adj_k<<<(NC * KS) / 256, gB, 0, stream>>>(kdcRe, kdcIm, iy, ix, wy, wx, tmpRe, tmpIm);
  cgemm512_k<<<gg, gb, 0, stream>>>(Ct, St, 1.0f, 0,
                                    tmpRe, tmpIm, 1.0f, (long long)GSZ,
                                    padRe, padIm, (long long)GSZ, 1.0f);
  cgemm512_k<<<gg, gb, 0, stream>>>(padRe, padIm, 1.0f, (long long)GSZ,
                                    Ct, St, 1.0f, 0,
                                    tmpRe, tmpIm, (long long)GSZ, 1.0f / (float)GRID);

  // final coil combine -> complex64 interleaved output
  combine_out_k<<<(IMG * IMG) / 256, gB, 0, stream>>>(tmpRe, tmpIm, sm_re, sm_im, sc, out);
}